// LocationAttention_3745211483034
// MI455X (gfx1250) — compile-verified
//
#include <hip/hip_runtime.h>
#include <hip/hip_bf16.h>
#include <math.h>

// ---------------- problem constants ----------------
static constexpr int B    = 128;
static constexpr int T    = 1024;
static constexpr int ENC  = 512;
static constexpr int QD   = 1024;
static constexpr int AD   = 128;   // ATT_DIM
static constexpr int F    = 32;    // ATT_FILTERS
static constexpr int KW   = 31;    // ATT_KERNEL
static constexpr int HALO = 15;    // (KW-1)/2, SAME padding

typedef __attribute__((ext_vector_type(2))) float v2f;
typedef __attribute__((ext_vector_type(4))) float v4f;
typedef __attribute__((ext_vector_type(8))) float v8f;

// =====================================================================
// Kernel 1: pq = query(128x1024) @ W_query(1024x128)  via f32 WMMA
// 64 waves, one 16x16 C-tile per wave, K loop in steps of 4.
// =====================================================================
__global__ __launch_bounds__(256) void la_pq_kernel(
    const float* __restrict__ q, const float* __restrict__ Wq,
    float* __restrict__ pq)
{
    const int wave = (blockIdx.x * blockDim.x + threadIdx.x) >> 5;  // 0..63
    const int lane = threadIdx.x & 31;
    const int m0 = (wave >> 3) << 4;        // 8 row tiles of 16
    const int n0 = (wave & 7) << 4;         // 8 col tiles of 16
    const int half = lane >> 4;             // 0 -> K{0,1}, 1 -> K{2,3}
    const int l = lane & 15;

    v8f c = {};
    for (int k0 = 0; k0 < QD; k0 += 4) {
        const int ka = k0 + half * 2;
        v2f a, b;
        a.x = q[(m0 + l) * QD + ka];
        a.y = q[(m0 + l) * QD + ka + 1];
        b.x = Wq[ka * AD + n0 + l];
        b.y = Wq[(ka + 1) * AD + n0 + l];
        c = __builtin_amdgcn_wmma_f32_16x16x4_f32(
                false, a, false, b, (short)0, c, false, false);
    }
    // C layout: VGPR r -> row (m0 + r + half*8), col (n0 + l)
#pragma unroll
    for (int r = 0; r < 8; ++r)
        pq[(m0 + r + half * 8) * AD + n0 + l] = c[r];
}

// =====================================================================
// Kernel 2: per (b, 16-t tile):
//   conv1d(aw_cat) -> loc(16x32) in LDS
//   loc @ W_loc (WMMA, 8 waves = 8 N-tiles of 16) + pq + pm
//   tanh, dot with v, cross-lane + LDS reduce -> energies(b, t)
// =====================================================================
__global__ __launch_bounds__(256) void la_energy_kernel(
    const float* __restrict__ aw,  const float* __restrict__ ck,
    const float* __restrict__ Wl,  const float* __restrict__ pq,
    const float* __restrict__ pm,  const float* __restrict__ v,
    float* __restrict__ energies)
{
    __shared__ float s_aw[16 + KW - 1][2];   // halo'd input rows
    __shared__ float s_ck[KW * 2 * F];       // conv kernel [k][c][f]
    __shared__ float s_wl[F][AD];            // W_loc 32x128
    __shared__ float s_loc[16][F];           // conv output tile
    __shared__ float s_pq[AD];
    __shared__ float s_v[AD];
    __shared__ float s_e[16];

    const int b  = blockIdx.x >> 6;          // T/16 = 64 tiles per batch
    const int t0 = (blockIdx.x & 63) << 4;
    const int tid = threadIdx.x;

    for (int i = tid; i < KW * 2 * F; i += 256) s_ck[i] = ck[i];
    for (int i = tid; i < F * AD;     i += 256) s_wl[i >> 7][i & (AD - 1)] = Wl[i];
    if (tid < AD) { s_pq[tid] = pq[b * AD + tid]; s_v[tid] = v[tid]; }
    if (tid < 2 * (16 + KW - 1)) {
        const int row = tid >> 1, cch = tid & 1;
        const int t = t0 + row - HALO;
        s_aw[row][cch] = (t >= 0 && t < T) ? aw[((size_t)b * T + t) * 2 + cch] : 0.f;
    }
    if (tid < 16) s_e[tid] = 0.f;
    __syncthreads();

    // conv: 16*32 = 512 outputs, 2 per thread
    for (int i = tid; i < 16 * F; i += 256) {
        const int row = i >> 5, f = i & 31;
        float acc = 0.f;
#pragma unroll
        for (int k = 0; k < KW; ++k) {
            acc = fmaf(s_aw[row + k][0], s_ck[(k * 2 + 0) * F + f], acc);
            acc = fmaf(s_aw[row + k][1], s_ck[(k * 2 + 1) * F + f], acc);
        }
        s_loc[row][f] = acc;
    }
    __syncthreads();

    // WMMA: wave w owns columns a = [16w, 16w+16)
    const int wave = tid >> 5, lane = tid & 31;
    const int n0 = wave << 4;
    const int half = lane >> 4;
    const int l = lane & 15;

    v8f c = {};
#pragma unroll
    for (int k0 = 0; k0 < F; k0 += 4) {
        const int ka = k0 + half * 2;
        v2f a, bb;
        a.x  = s_loc[l][ka];
        a.y  = s_loc[l][ka + 1];
        bb.x = s_wl[ka][n0 + l];
        bb.y = s_wl[ka + 1][n0 + l];
        c = __builtin_amdgcn_wmma_f32_16x16x4_f32(
                false, a, false, bb, (short)0, c, false, false);
    }

    const int a_col = n0 + l;
    const float vv = s_v[a_col];
    const float pqv = s_pq[a_col];
#pragma unroll
    for (int r = 0; r < 8; ++r) {
        const int row = r + half * 8;             // t within tile
        const size_t t = (size_t)(t0 + row);
        float e = c[r] + pqv +
                  __builtin_nontemporal_load(pm + ((size_t)b * T + t) * AD + a_col);
        float tv = tanhf(e) * vv;
        // reduce over the 16 N-lanes of this half-wave (masks 1..8 keep bit4)
#pragma unroll
        for (int off = 8; off >= 1; off >>= 1)
            tv += __shfl_xor(tv, off, 32);
        if (l == 0) atomicAdd(&s_e[row], tv);     // ds_add_f32, 8 waves combine
    }
    __syncthreads();
    if (tid < 16) energies[(size_t)b * T + t0 + tid] = s_e[tid];
}

// =====================================================================
// Kernel 3: row softmax over T=1024, one block per batch row
// =====================================================================
__global__ __launch_bounds__(256) void la_softmax_kernel(
    const float* __restrict__ e, float* __restrict__ w)
{
    __shared__ float red[8];
    const int b = blockIdx.x;
    const int tid = threadIdx.x;
    const int lane = tid & 31, wave = tid >> 5;

    float ev[4];
    float m = -3.4e38f;
#pragma unroll
    for (int i = 0; i < 4; ++i) {
        ev[i] = e[(size_t)b * T + tid + i * 256];
        m = fmaxf(m, ev[i]);
    }
#pragma unroll
    for (int off = 16; off >= 1; off >>= 1) m = fmaxf(m, __shfl_xor(m, off, 32));
    if (lane == 0) red[wave] = m;
    __syncthreads();
    float bm = red[0];
#pragma unroll
    for (int i = 1; i < 8; ++i) bm = fmaxf(bm, red[i]);
    __syncthreads();

    float s = 0.f;
#pragma unroll
    for (int i = 0; i < 4; ++i) { ev[i] = __expf(ev[i] - bm); s += ev[i]; }
#pragma unroll
    for (int off = 16; off >= 1; off >>= 1) s += __shfl_xor(s, off, 32);
    if (lane == 0) red[wave] = s;
    __syncthreads();
    float tot = 0.f;
#pragma unroll
    for (int i = 0; i < 8; ++i) tot += red[i];
    const float inv = 1.0f / tot;
#pragma unroll
    for (int i = 0; i < 4; ++i)
        w[(size_t)b * T + tid + i * 256] = ev[i] * inv;
}

// =====================================================================
// Kernel 4a: zero the context region of d_out (needed for atomics)
// =====================================================================
__global__ __launch_bounds__(256) void la_zero_kernel(float* __restrict__ p)
{
    p[blockIdx.x * 256 + threadIdx.x] = 0.f;
}

// =====================================================================
// Kernel 4b: context[b,d] = sum_t w[b,t] * mem[b,t,d]
// Streams `memory` (256 MB) once with B128 nontemporal loads.
// grid = (T/64 t-splits, B), 128 threads = 512 floats (v4f each).
// =====================================================================
__global__ __launch_bounds__(128) void la_context_kernel(
    const float* __restrict__ wts, const float* __restrict__ mem,
    float* __restrict__ ctx)
{
    __shared__ float s_w[64];
    const int b  = blockIdx.y;
    const int t0 = blockIdx.x * 64;
    const int tid = threadIdx.x;

    if (tid < 64) s_w[tid] = wts[(size_t)b * T + t0 + tid];
    __syncthreads();

    const v4f* m4 = reinterpret_cast<const v4f*>(
        mem + ((size_t)b * T + t0) * ENC) + tid;
    v4f acc = {0.f, 0.f, 0.f, 0.f};
#pragma unroll 4
    for (int i = 0; i < 64; ++i) {
        const v4f mv = __builtin_nontemporal_load(m4 + (size_t)i * (ENC / 4));
        const float w = s_w[i];
        acc.x = fmaf(w, mv.x, acc.x);
        acc.y = fmaf(w, mv.y, acc.y);
        acc.z = fmaf(w, mv.z, acc.z);
        acc.w = fmaf(w, mv.w, acc.w);
    }
    float* out = ctx + (size_t)b * ENC + tid * 4;
    atomicAdd(out + 0, acc.x);
    atomicAdd(out + 1, acc.y);
    atomicAdd(out + 2, acc.z);
    atomicAdd(out + 3, acc.w);
}

// =====================================================================
extern "C" void kernel_launch(void* const* d_in, const int* in_sizes, int n_in,
                              void* d_out, int out_size, void* d_ws, size_t ws_size,
                              hipStream_t stream)
{
    const float* query  = (const float*)d_in[0];   // (B, QD)
    const float* memory = (const float*)d_in[1];   // (B, T, ENC)
    const float* pm     = (const float*)d_in[2];   // (B, T, AD)
    const float* awc    = (const float*)d_in[3];   // (B, T, 2)
    const float* Wq     = (const float*)d_in[4];   // (QD, AD)
    const float* ck     = (const float*)d_in[5];   // (KW, 2, F)
    const float* Wl     = (const float*)d_in[6];   // (F, AD)
    const float* v      = (const float*)d_in[7];   // (AD, 1)

    float* ctx = (float*)d_out;                    // (B, ENC)
    float* wts = (float*)d_out + (size_t)B * ENC;  // (B, T)

    float* pq       = (float*)d_ws;                // B*AD floats
    float* energies = pq + B * AD;                 // B*T floats

    // 1) pq = query @ W_query   (WMMA f32 16x16x4)
    la_pq_kernel<<<8, 256, 0, stream>>>(query, Wq, pq);

    // 2) conv + loc@W_loc (WMMA) + tanh + dot v -> energies
    la_energy_kernel<<<B * (T / 16), 256, 0, stream>>>(
        awc, ck, Wl, pq, pm, v, energies);

    // 3) softmax over T -> attention weights (second output)
    la_softmax_kernel<<<B, 256, 0, stream>>>(energies, wts);

    // 4) context = weights . memory (first output)
    la_zero_kernel<<<(B * ENC) / 256, 256, 0, stream>>>(ctx);
    la_context_kernel<<<dim3(T / 64, B), 128, 0, stream>>>(wts, memory, ctx);
}